// RESCAL_59931973648702
// MI455X (gfx1250) — compile-verified
//
#include <hip/hip_runtime.h>

typedef __attribute__((ext_vector_type(2))) float v2f;
typedef __attribute__((ext_vector_type(8))) float v8f;

#define HDIM 128

// One workgroup (8 waves) per example. Wave a computes rows [16a,16a+16) of
// rt = R @ t via v_wmma_f32_16x16x4_f32 (fp32-exact), then dots with h.
__global__ __launch_bounds__(256) void rescal_score_kernel(
    const float* __restrict__ ent,     // [ENT_TOTAL, 128]
    const float* __restrict__ rel,     // [REL_TOTAL, 128*128]
    const int* __restrict__ pos_h, const int* __restrict__ pos_t,
    const int* __restrict__ pos_r,
    const int* __restrict__ neg_h, const int* __restrict__ neg_t,
    const int* __restrict__ neg_r,
    float* __restrict__ scores,        // [B + B*K]
    int B)
{
    const int n = blockIdx.x;
    int hIdx, tIdx, rIdx;
    if (n < B) {
        hIdx = pos_h[n]; tIdx = pos_t[n]; rIdx = pos_r[n];
    } else {
        const int m = n - B;
        hIdx = neg_h[m]; tIdx = neg_t[m]; rIdx = neg_r[m];
    }

    const float* __restrict__ hv = ent + (size_t)hIdx * HDIM;
    const float* __restrict__ tv = ent + (size_t)tIdx * HDIM;
    const float* __restrict__ R  = rel + (size_t)rIdx * (HDIM * HDIM);

    const int tid  = threadIdx.x;
    const int wave = tid >> 5;        // 0..7 : row-block of R
    const int lane = tid & 31;
    const int lo   = lane & 15;       // M within tile (for A) / N column (for B)
    const int half = lane >> 4;       // 0: K=0..1 in VGPR0..1, 1: K=2..3

    // A-fragment source: R[16*wave + lo, j0 + 2*half + {0,1}]
    const float* __restrict__ Rrow = R + (wave * 16 + lo) * HDIM + 2 * half;
    // B-fragment source: t[j0 + 2*half + {0,1}] (replicated across N columns)
    const float* __restrict__ tp = tv + 2 * half;

    v8f acc = {0.f, 0.f, 0.f, 0.f, 0.f, 0.f, 0.f, 0.f};

#pragma unroll 8
    for (int j0 = 0; j0 < HDIM; j0 += 4) {
        v2f a = *(const v2f*)(Rrow + j0);
        v2f b = *(const v2f*)(tp + j0);
        // D = A(16x4 f32) x B(4x16 f32) + C ; fp32-exact accumulate
        acc = __builtin_amdgcn_wmma_f32_16x16x4_f32(
            /*neg_a=*/false, a, /*neg_b=*/false, b,
            /*c_mod=*/(short)0, acc, /*reuse_a=*/false, /*reuse_b=*/false);
    }

    // acc[v] holds rt[16*wave + v + 8*half] (identical across the 16 lanes of
    // each half). Dot with matching slice of h.
    const float* __restrict__ hb = hv + wave * 16 + half * 8;
    float partial = 0.f;
#pragma unroll
    for (int v = 0; v < 8; ++v) partial += acc[v] * hb[v];

    __shared__ float s_part[16];
    if (lo == 0) s_part[wave * 2 + half] = partial;
    __syncthreads();

    if (tid == 0) {
        float s = 0.f;
#pragma unroll
        for (int i = 0; i < 16; ++i) s += s_part[i];
        scores[n] = s;
    }
}

// Single block: hinge over B positives with mean over K negatives, then a
// deterministic LDS tree reduction to the scalar loss.
__global__ __launch_bounds__(1024) void rescal_loss_kernel(
    const float* __restrict__ scores, float* __restrict__ out,
    int B, int K, float margin)
{
    __shared__ float red[1024];
    const int b = threadIdx.x;

    float hinge = 0.f;
    if (b < B) {
        const float* __restrict__ neg = scores + B + (size_t)b * K;
        float ns = 0.f;
        for (int k = 0; k < K; ++k) ns += neg[k];
        ns *= (1.0f / (float)K);
        const float v = ns - scores[b] + margin;
        hinge = v > 0.f ? v : 0.f;
    }
    red[b] = hinge;
    __syncthreads();

    for (int s = 512; s > 0; s >>= 1) {
        if (b < s) red[b] += red[b + s];
        __syncthreads();
    }
    if (b == 0) out[0] = red[0];
}

extern "C" void kernel_launch(void* const* d_in, const int* in_sizes, int n_in,
                              void* d_out, int out_size, void* d_ws, size_t ws_size,
                              hipStream_t stream) {
    const float* ent   = (const float*)d_in[0];
    const float* rel   = (const float*)d_in[1];
    const int*   pos_h = (const int*)d_in[2];
    const int*   pos_t = (const int*)d_in[3];
    const int*   pos_r = (const int*)d_in[4];
    const int*   neg_h = (const int*)d_in[5];
    const int*   neg_t = (const int*)d_in[6];
    const int*   neg_r = (const int*)d_in[7];

    const int B  = in_sizes[2];        // 1024
    const int NK = in_sizes[5];        // B*K = 10240
    const int K  = NK / B;             // 10
    const int NT = B + NK;             // 11264 examples

    float* scores = (float*)d_ws;      // NT floats of scratch

    rescal_score_kernel<<<NT, 256, 0, stream>>>(
        ent, rel, pos_h, pos_t, pos_r, neg_h, neg_t, neg_r, scores, B);

    rescal_loss_kernel<<<1, 1024, 0, stream>>>(
        scores, (float*)d_out, B, K, 1.0f);
}